// BiLSTMCRF_652835029265
// MI455X (gfx1250) — compile-verified
//
#include <hip/hip_runtime.h>

// BiLSTM-CRF for MI455X (gfx1250), fp32 WMMA (V_WMMA_F32_16X16X4_F32) throughout.
// Workspace layout (floats): XpF[B*S*256] | XpB[B*S*256] | hs[B*S*128] | logitsP[B*S*32]
// total ~672 MB of d_ws.

#define B_ 512
#define S_ 512
#define E_ 100
#define HD_ 64
#define H_ 128
#define T_ 25
#define TP_ 32      // tag dim padded to 32 for WMMA N-tiles
#define G_ 256      // 4*HD gate width
#define START_ 1
#define STOP_ 2

typedef float v2f __attribute__((ext_vector_type(2)));
typedef float v8f __attribute__((ext_vector_type(8)));

__device__ __forceinline__ v8f wmma_f32_k4(v2f a, v2f b, v8f c) {
  // D = A(16x4) * B(4x16) + C(16x16), fp32
  return __builtin_amdgcn_wmma_f32_16x16x4_f32(false, a, false, b, (short)0, c,
                                               false, false);
}

// ---------------------------------------------------------------------------
// Kernel 1: embedding gather + input projection GEMM (+ fused biases).
// Xp[dir][(b*S+s)][g] = sum_e embed[sent[b,s]][e] * W_ih[dir][g][e] + b_ih[g]+b_hh[g]
// Workgroup: 64 output rows x 256 gates, 8 waves each owning two 16-wide N-tiles.
// ---------------------------------------------------------------------------
#define XP_ESTR 106  // LDS row stride for gathered embeddings (bank-conflict pad)
__global__ __launch_bounds__(256) void xproj_kernel(
    const int* __restrict__ sent, const float* __restrict__ embed,
    const float* __restrict__ WihF, const float* __restrict__ bihF,
    const float* __restrict__ bhhF, const float* __restrict__ WihB,
    const float* __restrict__ bihB, const float* __restrict__ bhhB,
    float* __restrict__ XpF, float* __restrict__ XpB) {
  __shared__ float eL[64 * XP_ESTR];
  const int dir = blockIdx.y;
  const float* __restrict__ Wih = dir ? WihB : WihF;
  const float* __restrict__ bih = dir ? bihB : bihF;
  const float* __restrict__ bhh = dir ? bhhB : bhhF;
  float* __restrict__ Xp = dir ? XpB : XpF;
  const int mBase = blockIdx.x * 64;

  {  // gather 64 embedding rows into LDS (4 threads per row, 25 cols each)
    int r = threadIdx.x >> 2;
    int part = threadIdx.x & 3;
    int tok = sent[mBase + r];
    const float* src = embed + (size_t)tok * E_;
    float* dst = eL + r * XP_ESTR;
    for (int c = part * 25; c < part * 25 + 25; ++c) dst[c] = src[c];
  }
  __syncthreads();

  const int lane = threadIdx.x & 31;
  const int wave = threadIdx.x >> 5;
  const int nb0 = wave * 32;  // this wave's gate columns [nb0, nb0+32)
  const int ln = lane & 15;
  const int kh = lane >> 4;
  const float bs0 = bih[nb0 + ln] + bhh[nb0 + ln];
  const float bs1 = bih[nb0 + 16 + ln] + bhh[nb0 + 16 + ln];

  for (int msub = 0; msub < 4; ++msub) {
    const int m0 = msub * 16;
    v8f acc0, acc1;
#pragma unroll
    for (int r = 0; r < 8; ++r) { acc0[r] = bs0; acc1[r] = bs1; }
#pragma unroll
    for (int ks = 0; ks < 25; ++ks) {  // K = 100 = 25 * 4
      const int col = 4 * ks + 2 * kh;
      v2f a  = *(const v2f*)(eL + (m0 + ln) * XP_ESTR + col);
      v2f b0 = *(const v2f*)(Wih + (size_t)(nb0 + ln) * E_ + col);
      v2f b1 = *(const v2f*)(Wih + (size_t)(nb0 + 16 + ln) * E_ + col);
      acc0 = wmma_f32_k4(a, b0, acc0);
      acc1 = wmma_f32_k4(a, b1, acc1);
    }
    const size_t rowBase = (size_t)(mBase + m0);
#pragma unroll
    for (int r = 0; r < 8; ++r) {
      const int m = r + 8 * kh;
      Xp[(rowBase + m) * G_ + nb0 + ln] = acc0[r];
      Xp[(rowBase + m) * G_ + nb0 + 16 + ln] = acc1[r];
    }
  }
}

// ---------------------------------------------------------------------------
// Kernel 2: LSTM recurrence, one workgroup per (batch-tile of 16, direction).
// h kept in LDS, c in registers, W_hh B-fragments resident in VGPRs.
// Per step: z = Xp_tile + h @ W_hh^T via 16 chained fp32 WMMAs per N-tile.
// ---------------------------------------------------------------------------
#define HL_STR 66
#define ZL_STR 258
__global__ __launch_bounds__(256) void lstm_kernel(
    const float* __restrict__ WhhF, const float* __restrict__ WhhB,
    const float* __restrict__ h0, const float* __restrict__ c0,
    const float* __restrict__ XpF, const float* __restrict__ XpB,
    float* __restrict__ hs) {
  __shared__ float hL[16 * HL_STR];
  __shared__ float zL[16 * ZL_STR];
  const int dir = blockIdx.y;
  const float* __restrict__ Whh = dir ? WhhB : WhhF;
  const float* __restrict__ Xp = dir ? XpB : XpF;
  const int bBase = blockIdx.x * 16;
  const int lane = threadIdx.x & 31;
  const int wave = threadIdx.x >> 5;
  const int ln = lane & 15, kh = lane >> 4;
  const int nb0 = wave * 32;

  // Preload W_hh fragments (B matrices) for this wave's two N-tiles: 64 VGPRs.
  v2f bw0[16], bw1[16];
#pragma unroll
  for (int ks = 0; ks < 16; ++ks) {
    const int col = 4 * ks + 2 * kh;
    bw0[ks] = *(const v2f*)(Whh + (size_t)(nb0 + ln) * HD_ + col);
    bw1[ks] = *(const v2f*)(Whh + (size_t)(nb0 + 16 + ln) * HD_ + col);
  }
  // initial hidden state -> LDS
  for (int idx = threadIdx.x; idx < 16 * HD_; idx += 256) {
    const int m = idx >> 6, j = idx & 63;
    hL[m * HL_STR + j] = h0[((size_t)dir * B_ + bBase + m) * HD_ + j];
  }
  // cell state in registers: thread t owns (m = t>>4, j = 4*(t&15)+q)
  const int cm = threadIdx.x >> 4;
  const int cj0 = (threadIdx.x & 15) * 4;
  float cReg[4];
#pragma unroll
  for (int q = 0; q < 4; ++q)
    cReg[q] = c0[((size_t)dir * B_ + bBase + cm) * HD_ + cj0 + q];
  __syncthreads();

  for (int s = 0; s < S_; ++s) {
    const int sIdx = dir ? (S_ - 1 - s) : s;
    // C initialized with precomputed x-projection (biases already folded in)
    v8f acc0, acc1;
#pragma unroll
    for (int r = 0; r < 8; ++r) {
      const int m = r + 8 * kh;
      const size_t row = ((size_t)(bBase + m) * S_ + sIdx) * G_;
      acc0[r] = Xp[row + nb0 + ln];
      acc1[r] = Xp[row + nb0 + 16 + ln];
    }
    if (s + 1 < S_) {  // prefetch next timestep's x-projection tile
      const int sn = dir ? (S_ - 2 - s) : (s + 1);
      __builtin_prefetch(
          Xp + ((size_t)(bBase + ln) * S_ + sn) * G_ + nb0 + kh * 16, 0, 0);
    }
#pragma unroll
    for (int ks = 0; ks < 16; ++ks) {  // K = HD = 64 = 16 * 4
      const int col = 4 * ks + 2 * kh;
      v2f a = *(const v2f*)(hL + ln * HL_STR + col);
      acc0 = wmma_f32_k4(a, bw0[ks], acc0);
      acc1 = wmma_f32_k4(a, bw1[ks], acc1);
    }
#pragma unroll
    for (int r = 0; r < 8; ++r) {
      const int m = r + 8 * kh;
      zL[m * ZL_STR + nb0 + ln] = acc0[r];
      zL[m * ZL_STR + nb0 + 16 + ln] = acc1[r];
    }
    __syncthreads();
    // gate nonlinearity + state update (torch gate order i, f, g, o)
    {
      const float* zRow = zL + cm * ZL_STR;
      const size_t hsRow = ((size_t)(bBase + cm) * S_ + sIdx) * H_ + dir * HD_;
#pragma unroll
      for (int q = 0; q < 4; ++q) {
        const int j = cj0 + q;
        const float zi = zRow[j];
        const float zf = zRow[HD_ + j];
        const float zg = zRow[2 * HD_ + j];
        const float zo = zRow[3 * HD_ + j];
        const float si = 1.f / (1.f + __expf(-zi));
        const float sf = 1.f / (1.f + __expf(-zf));
        const float so = 1.f / (1.f + __expf(-zo));
        const float tg = 1.f - 2.f / (__expf(2.f * zg) + 1.f);
        const float cc = sf * cReg[q] + si * tg;
        cReg[q] = cc;
        const float hh = so * (1.f - 2.f / (__expf(2.f * cc) + 1.f));
        hL[cm * HL_STR + j] = hh;
        hs[hsRow + j] = hh;
      }
    }
    __syncthreads();
  }
}

// ---------------------------------------------------------------------------
// Kernel 3: output projection: logits[b,s,t] = hs[b,s,:] @ W_out[t,:] + b_out[t]
// N padded 25 -> 32 (two N-tiles), stored as [B*S][32].
// ---------------------------------------------------------------------------
#define WO_STR 130
__global__ __launch_bounds__(256) void logits_kernel(
    const float* __restrict__ hs, const float* __restrict__ Wout,
    const float* __restrict__ bout, float* __restrict__ logitsP) {
  __shared__ float wL[TP_ * WO_STR];
  __shared__ float bL[TP_];
  for (int idx = threadIdx.x; idx < TP_ * H_; idx += 256) {
    const int r = idx >> 7, c = idx & 127;
    wL[r * WO_STR + c] = (r < T_) ? Wout[r * H_ + c] : 0.f;
  }
  if (threadIdx.x < TP_)
    bL[threadIdx.x] = (threadIdx.x < T_) ? bout[threadIdx.x] : 0.f;
  __syncthreads();

  const int lane = threadIdx.x & 31;
  const int wave = threadIdx.x >> 5;
  const int ln = lane & 15, kh = lane >> 4;
  const size_t rowBase = (size_t)blockIdx.x * 128 + wave * 16;
  const float bs0 = bL[ln], bs1 = bL[16 + ln];
  v8f acc0, acc1;
#pragma unroll
  for (int r = 0; r < 8; ++r) { acc0[r] = bs0; acc1[r] = bs1; }
#pragma unroll
  for (int ks = 0; ks < 32; ++ks) {  // K = H = 128 = 32 * 4
    const int col = 4 * ks + 2 * kh;
    v2f a  = *(const v2f*)(hs + (rowBase + ln) * H_ + col);
    v2f b0 = *(const v2f*)(wL + ln * WO_STR + col);
    v2f b1 = *(const v2f*)(wL + (16 + ln) * WO_STR + col);
    acc0 = wmma_f32_k4(a, b0, acc0);
    acc1 = wmma_f32_k4(a, b1, acc1);
  }
#pragma unroll
  for (int r = 0; r < 8; ++r) {
    const int m = r + 8 * kh;
    logitsP[(rowBase + m) * TP_ + ln] = acc0[r];
    logitsP[(rowBase + m) * TP_ + 16 + ln] = acc1[r];
  }
}

// ---------------------------------------------------------------------------
// Kernel 4: CRF gold score + forward algorithm; one wave32 per batch element.
// Lane j owns tag j; trans column cached in 25 VGPRs; prev broadcast via shfl.
// Accumulates sum(total - real) into d_out with atomicAdd.
// ---------------------------------------------------------------------------
__global__ __launch_bounds__(256) void crf_kernel(
    const int* __restrict__ tags, const float* __restrict__ trans,
    const float* __restrict__ logitsP, float* __restrict__ out) {
  const int lane = threadIdx.x & 31;
  const int wave = threadIdx.x >> 5;
  const int b = blockIdx.x * 8 + wave;
  const int j = lane;

  float trc[T_];
#pragma unroll
  for (int i = 0; i < T_; ++i) trc[i] = (j < T_) ? trans[i * T_ + j] : 0.f;

  const int* __restrict__ btags = tags + (size_t)b * S_;
  const float* __restrict__ blog = logitsP + (size_t)b * S_ * TP_;

  // gold path score, lane-parallel over time
  float em = 0.f, trg = 0.f;
  for (int s = lane; s < S_; s += 32) {
    const int cur = btags[s];
    const int pre = (s == 0) ? START_ : btags[s - 1];
    em += blog[(size_t)s * TP_ + cur];
    trg += trans[pre * T_ + cur];
  }
  float gold = em + trg;
#pragma unroll
  for (int off = 16; off > 0; off >>= 1) gold += __shfl_xor(gold, off, 32);

  // forward recursion: prev_j <- logsumexp_i(prev_i + trans[i][j]) + logit_j
  float prev = (j < T_) ? 0.f : -1e30f;
  for (int s = 0; s < S_; ++s) {
    const float lt = blog[(size_t)s * TP_ + j];
    float mx = -1e30f;
#pragma unroll
    for (int i = 0; i < T_; ++i)
      mx = fmaxf(mx, __shfl(prev, i, 32) + trc[i]);
    float sm = 0.f;
#pragma unroll
    for (int i = 0; i < T_; ++i)
      sm += __expf(__shfl(prev, i, 32) + trc[i] - mx);
    float nv = mx + __logf(sm) + lt;
    prev = (j < T_) ? nv : -1e30f;
  }
  // total = logsumexp_j(prev_j + trans[j][STOP])
  const float fin = (j < T_) ? (prev + trans[j * T_ + STOP_]) : -1e30f;
  float mx = -1e30f;
#pragma unroll
  for (int i = 0; i < T_; ++i) mx = fmaxf(mx, __shfl(fin, i, 32));
  float sm = 0.f;
#pragma unroll
  for (int i = 0; i < T_; ++i) sm += __expf(__shfl(fin, i, 32) - mx);
  const float total = mx + __logf(sm);

  if (lane == 0) {
    const float gfin = gold + trans[btags[S_ - 1] * T_ + STOP_];
    atomicAdd(out, total - gfin);
  }
}

// ---------------------------------------------------------------------------
extern "C" void kernel_launch(void* const* d_in, const int* in_sizes, int n_in,
                              void* d_out, int out_size, void* d_ws,
                              size_t ws_size, hipStream_t stream) {
  const int*   sent  = (const int*)d_in[0];
  const int*   tags  = (const int*)d_in[1];
  const float* embed = (const float*)d_in[2];
  const float* WihF  = (const float*)d_in[3];
  const float* WhhF  = (const float*)d_in[4];
  const float* bihF  = (const float*)d_in[5];
  const float* bhhF  = (const float*)d_in[6];
  const float* WihB  = (const float*)d_in[7];
  const float* WhhB  = (const float*)d_in[8];
  const float* bihB  = (const float*)d_in[9];
  const float* bhhB  = (const float*)d_in[10];
  const float* h0    = (const float*)d_in[11];
  const float* c0    = (const float*)d_in[12];
  const float* Wout  = (const float*)d_in[13];
  const float* bout  = (const float*)d_in[14];
  const float* trans = (const float*)d_in[15];

  float* ws = (float*)d_ws;
  const size_t nBS = (size_t)B_ * S_;
  float* XpF = ws;
  float* XpB = XpF + nBS * G_;
  float* hsb = XpB + nBS * G_;
  float* logitsP = hsb + nBS * H_;

  hipMemsetAsync(d_out, 0, sizeof(float), stream);

  dim3 gX((unsigned)(nBS / 64), 2);
  xproj_kernel<<<gX, 256, 0, stream>>>(sent, embed, WihF, bihF, bhhF, WihB,
                                       bihB, bhhB, XpF, XpB);
  dim3 gL(B_ / 16, 2);
  lstm_kernel<<<gL, 256, 0, stream>>>(WhhF, WhhB, h0, c0, XpF, XpB, hsb);
  logits_kernel<<<(unsigned)(nBS / 128), 256, 0, stream>>>(hsb, Wout, bout,
                                                           logitsP);
  crf_kernel<<<B_ / 8, 256, 0, stream>>>(tags, trans, logitsP, (float*)d_out);
}